// GraphEncoder_68753836474798
// MI455X (gfx1250) — compile-verified
//
#include <hip/hip_runtime.h>

typedef __attribute__((ext_vector_type(2))) float v2f;
typedef __attribute__((ext_vector_type(8))) float v8f;

#define F_IN 10
#define F1   128
#define F2   256
#define LAT  64
#define NT   4      // 16-wide column tiles per wave -> 16x64 output per wave

// ---------------- degree / normalization ----------------
__global__ void k_deg_init(float* deg, int n) {
    int i = blockIdx.x * blockDim.x + threadIdx.x;
    if (i < n) deg[i] = 1.0f;                       // self-loop
}
__global__ void k_deg_count(const int* __restrict__ dstv, float* deg, int e) {
    int i = blockIdx.x * blockDim.x + threadIdx.x;
    if (i < e) atomicAdd(&deg[dstv[i]], 1.0f);
}
__global__ void k_deg_rsqrt(float* deg, int n) {
    int i = blockIdx.x * blockDim.x + threadIdx.x;
    if (i < n) deg[i] = rsqrtf(deg[i]);             // deg >= 1 always
}

// ---- fp32 WMMA GEMM: C[M,Fout] = A[M,K] * B[K,Fout] (+bias) ----
// One wave computes a 16x(16*NT) output slab with V_WMMA_F32_16X16X4_F32.
// Per K-step: issue ALL loads (1x b64 A + 2*NT x b32 B) into distinct regs,
// one s_wait_loadcnt, then NT back-to-back WMMAs on independent acc chains.
__global__ void k_gemm_f32_wmma(const float* __restrict__ A,
                                const float* __restrict__ B,
                                const float* __restrict__ bias,
                                float* __restrict__ C,
                                int M, int K, int lda, int ldb, int ldc) {
    const int lane = threadIdx.x & 31;
    const int hi   = lane >> 4;        // 0: lanes 0-15, 1: lanes 16-31
    const int lr   = lane & 15;
    const int m0   = blockIdx.y << 4;
    const int n0   = blockIdx.x * (16 * NT);
    int arow = m0 + lr;
    if (arow >= M) arow = M - 1;       // clamp; bad rows never stored
    const int bcol = n0 + lr;

    v8f acc[NT];
#pragma unroll
    for (int t = 0; t < NT; ++t) acc[t] = (v8f){};

    const float* Ap = A + arow * lda;  // this lane's A row
    const int K4 = K & ~3;

    int k = 0;
#pragma unroll 2
    for (; k < K4; k += 4) {
        const int ka = k + (hi << 1);               // this half-wave's K pair
        const v2f a = *(const v2f*)(Ap + ka);       // global_load_b64
        const float* B0 = B + ka * ldb + bcol;
        const float* B1 = B0 + ldb;
        v2f b[NT];                                  // distinct regs: loads first,
#pragma unroll
        for (int t = 0; t < NT; ++t) {              // then one wait, then WMMAs
            b[t].x = B0[16 * t];
            b[t].y = B1[16 * t];
        }
#pragma unroll
        for (int t = 0; t < NT; ++t) {
            acc[t] = __builtin_amdgcn_wmma_f32_16x16x4_f32(
                false, a, false, b[t], (short)0, acc[t], false, false);
        }
    }
    if (k < K) {                                    // runs once, only if K%4 != 0
        const int ka = k + (hi << 1);
        v2f a;
        a.x = (ka     < K) ? Ap[ka]     : 0.0f;
        a.y = (ka + 1 < K) ? Ap[ka + 1] : 0.0f;
        v2f b[NT];
#pragma unroll
        for (int t = 0; t < NT; ++t) {
            b[t].x = (ka     < K) ? B[(ka    ) * ldb + bcol + 16 * t] : 0.0f;
            b[t].y = (ka + 1 < K) ? B[(ka + 1) * ldb + bcol + 16 * t] : 0.0f;
        }
#pragma unroll
        for (int t = 0; t < NT; ++t) {
            acc[t] = __builtin_amdgcn_wmma_f32_16x16x4_f32(
                false, a, false, b[t], (short)0, acc[t], false, false);
        }
    }

#pragma unroll
    for (int t = 0; t < NT; ++t) {
        const int col = bcol + 16 * t;
        const float bv = bias ? bias[col] : 0.0f;
#pragma unroll
        for (int v = 0; v < 8; ++v) {
            const int row = m0 + (hi << 3) + v;     // VGPR v: M = v (+8 upper half)
            if (row < M) C[row * ldc + col] = acc[t][v] + bv;
        }
    }
}

// ---------------- aggregation ----------------
// agg[i,f] = hW[i,f] * dis[i]^2   (self-loop term, also overwrites old input)
__global__ void k_self_init(const float* __restrict__ hW, const float* __restrict__ dis,
                            float* __restrict__ agg, int total, int shift) {
    int gid = blockIdx.x * blockDim.x + threadIdx.x;
    if (gid < total) {
        const float s = dis[gid >> shift];
        agg[gid] = hW[gid] * s * s;
    }
}
// one thread per (edge, 4 features); F = 1<<shift floats per row
__global__ void k_scatter4(const float* __restrict__ hW,
                           const int* __restrict__ srcv, const int* __restrict__ dstv,
                           const float* __restrict__ dis, float* __restrict__ agg,
                           int e, int shift) {
    int gid = blockIdx.x * blockDim.x + threadIdx.x;
    const int q4   = shift - 2;                     // quads per row = 1<<q4
    const int f4   = gid & ((1 << q4) - 1);
    const int ed   = gid >> q4;
    if (ed < e) {
        const int s = srcv[ed], d = dstv[ed];
        const float w = dis[s] * dis[d];
        const float4 h = *(const float4*)(hW + (s << shift) + (f4 << 2));
        float* out = agg + (d << shift) + (f4 << 2);
        atomicAdd(out + 0, h.x * w);
        atomicAdd(out + 1, h.y * w);
        atomicAdd(out + 2, h.z * w);
        atomicAdd(out + 3, h.w * w);
    }
}
__global__ void k_bias_relu(float* __restrict__ h, const float* __restrict__ b,
                            int total, int mask) {
    int gid = blockIdx.x * blockDim.x + threadIdx.x;
    if (gid < total) h[gid] = fmaxf(h[gid] + b[gid & mask], 0.0f);
}

// ---------------- pooling ----------------
__global__ void k_zero(float* p, int n) {
    int i = blockIdx.x * blockDim.x + threadIdx.x;
    if (i < n) p[i] = 0.0f;
}
// one thread per (node, 4 features of 256)
__global__ void k_pool_sum4(const float* __restrict__ h, const int* __restrict__ batch,
                            float* __restrict__ sums, float* __restrict__ cnt, int n) {
    int gid = blockIdx.x * blockDim.x + threadIdx.x;
    const int i = gid >> 6, f4 = gid & 63;
    if (i < n) {
        const int g = batch[i];
        const float4 v = *(const float4*)(h + (i << 8) + (f4 << 2));
        float* out = sums + (g << 8) + (f4 << 2);
        atomicAdd(out + 0, v.x);
        atomicAdd(out + 1, v.y);
        atomicAdd(out + 2, v.z);
        atomicAdd(out + 3, v.w);
        if (f4 == 0) atomicAdd(&cnt[g], 1.0f);
    }
}
__global__ void k_pool_div(float* __restrict__ sums, const float* __restrict__ cnt, int total) {
    int gid = blockIdx.x * blockDim.x + threadIdx.x;
    if (gid < total) sums[gid] /= fmaxf(cnt[gid >> 8], 1.0f);
}

static inline int cdiv(long long a, int b) { return (int)((a + b - 1) / b); }

extern "C" void kernel_launch(void* const* d_in, const int* in_sizes, int n_in,
                              void* d_out, int out_size, void* d_ws, size_t ws_size,
                              hipStream_t stream) {
    (void)n_in; (void)ws_size;
    const float* x    = (const float*)d_in[0];
    const int*   ei   = (const int*)  d_in[1];
    const int*   bat  = (const int*)  d_in[2];
    const float* W1   = (const float*)d_in[3];  const float* b1  = (const float*)d_in[4];
    const float* W2   = (const float*)d_in[5];  const float* b2  = (const float*)d_in[6];
    const float* W3   = (const float*)d_in[7];  const float* b3  = (const float*)d_in[8];
    const float* Wmu  = (const float*)d_in[9];  const float* bmu = (const float*)d_in[10];
    const float* Wlv  = (const float*)d_in[11]; const float* blv = (const float*)d_in[12];

    const int N = in_sizes[0] / F_IN;
    const int E = in_sizes[1] / 2;
    const int G = out_size / (2 * LAT);
    const int* srcv = ei;          // edge_index[0]
    const int* dstv = ei + E;      // edge_index[1]

    float* ws     = (float*)d_ws;
    float* dis    = ws;                              // N
    float* cnt    = dis + N;                         // G
    float* pooled = cnt + G;                         // G*256
    float* bufB   = pooled + (size_t)G * F2;         // N*256 (GEMM out)
    float* bufC   = bufB + (size_t)N * F2;           // N*256 (agg / next input)

    float* outmu = (float*)d_out;
    float* outlv = outmu + (size_t)G * LAT;

    const dim3 B256(256);

    // --- symmetric normalization: dis = rsqrt(1 + in-degree) ---
    k_deg_init <<<cdiv(N, 256), B256, 0, stream>>>(dis, N);
    k_deg_count<<<cdiv(E, 256), B256, 0, stream>>>(dstv, dis, E);
    k_deg_rsqrt<<<cdiv(N, 256), B256, 0, stream>>>(dis, N);

    // --- layer 1: 10 -> 128 ---
    k_gemm_f32_wmma<<<dim3(F1/(16*NT), cdiv(N,16)), 32, 0, stream>>>(x, W1, nullptr, bufB, N, F_IN, F_IN, F1, F1);
    k_self_init<<<cdiv((long long)N*F1, 256), B256, 0, stream>>>(bufB, dis, bufC, N*F1, 7);
    k_scatter4 <<<cdiv((long long)E*F1/4, 256), B256, 0, stream>>>(bufB, srcv, dstv, dis, bufC, E, 7);
    k_bias_relu<<<cdiv((long long)N*F1, 256), B256, 0, stream>>>(bufC, b1, N*F1, F1-1);

    // --- layer 2: 128 -> 256 ---
    k_gemm_f32_wmma<<<dim3(F2/(16*NT), cdiv(N,16)), 32, 0, stream>>>(bufC, W2, nullptr, bufB, N, F1, F1, F2, F2);
    k_self_init<<<cdiv((long long)N*F2, 256), B256, 0, stream>>>(bufB, dis, bufC, N*F2, 8);
    k_scatter4 <<<cdiv((long long)E*F2/4, 256), B256, 0, stream>>>(bufB, srcv, dstv, dis, bufC, E, 8);
    k_bias_relu<<<cdiv((long long)N*F2, 256), B256, 0, stream>>>(bufC, b2, N*F2, F2-1);

    // --- layer 3: 256 -> 256 ---
    k_gemm_f32_wmma<<<dim3(F2/(16*NT), cdiv(N,16)), 32, 0, stream>>>(bufC, W3, nullptr, bufB, N, F2, F2, F2, F2);
    k_self_init<<<cdiv((long long)N*F2, 256), B256, 0, stream>>>(bufB, dis, bufC, N*F2, 8);
    k_scatter4 <<<cdiv((long long)E*F2/4, 256), B256, 0, stream>>>(bufB, srcv, dstv, dis, bufC, E, 8);
    k_bias_relu<<<cdiv((long long)N*F2, 256), B256, 0, stream>>>(bufC, b3, N*F2, F2-1);

    // --- global mean pool ---
    k_zero<<<cdiv((long long)G*F2, 256), B256, 0, stream>>>(pooled, G*F2);
    k_zero<<<cdiv(G, 256), B256, 0, stream>>>(cnt, G);
    k_pool_sum4<<<cdiv((long long)N*F2/4, 256), B256, 0, stream>>>(bufC, bat, pooled, cnt, N);
    k_pool_div<<<cdiv((long long)G*F2, 256), B256, 0, stream>>>(pooled, cnt, G*F2);

    // --- heads: mu and logvar, bias fused, written straight to d_out ---
    k_gemm_f32_wmma<<<dim3(LAT/(16*NT), cdiv(G,16)), 32, 0, stream>>>(pooled, Wmu, bmu, outmu, G, F2, F2, LAT, LAT);
    k_gemm_f32_wmma<<<dim3(LAT/(16*NT), cdiv(G,16)), 32, 0, stream>>>(pooled, Wlv, blv, outlv, G, F2, F2, LAT, LAT);
}